// BlockSample_33930241638701
// MI455X (gfx1250) — compile-verified
//
#include <hip/hip_runtime.h>
#include <stdint.h>

// Problem dimensions (fixed by the reference)
#define NB   8
#define NC   192
#define NH   48
#define NW   48
#define CPB  48               // channels per block
#define NCG  (NC / CPB)       // 4 channel groups
#define RS   50               // LDS row stride in dwords: 200B rows (8B-aligned async b64
                              // dests) and gcd(50,64)=2 -> conflict-free 32-lane reads

#if defined(__AMDGCN__) && __has_builtin(__builtin_amdgcn_global_load_async_to_lds_b64) && \
    __has_builtin(__builtin_amdgcn_s_wait_asynccnt)
#define ASYNC_LDS 1
#else
#define ASYNC_LDS 0
#endif

// Async builtin prototype (confirmed by compiler diagnostics):
//   void __builtin_amdgcn_global_load_async_to_lds_b64(v2i AS1* src, v2i AS3* dst,
//                                                      imm int offset, imm int cpol)
typedef int v2i __attribute__((vector_size(8)));
typedef __attribute__((address_space(1))) v2i* as1_v2i;
typedef __attribute__((address_space(3))) v2i* as3_v2i;
typedef float vf4 __attribute__((ext_vector_type(4)));

__global__ __launch_bounds__(192) void onehot_tap_gather(const float* __restrict__ in,
                                                         float* __restrict__ out)
{
    __shared__ __align__(16) float tile[CPB * 4 * RS];   // 48ch x 4rows x 50dw = 38.4 KB

    const int tid = threadIdx.x;          // 192 threads = 6 waves
    const int cg  = blockIdx.x;           // channel group 0..3
    const int y   = blockIdx.y;           // output row 0..47
    const int b   = blockIdx.z;           // batch 0..7
    const int c0  = cg * CPB;

    const float* src = in + (size_t)b * NC * NH * NW;

    // ---- Stage: 48 ch x 4 rows (y-3..y) x 24 8-byte units -> LDS (async DMA) ----
    // Per thread: fixed (u,row), channel steps by 2 each iteration -> 24 async b64 loads.
    {
        const int u  = tid % 24;          // 8-byte unit within row
        const int rw = (tid / 24) & 3;    // tap row 0..3  (input row y+rw-3)
        const int r  = y + rw - 3;
        if (r >= 0) {                     // r <= 47 always; skip above-top rows
            const int ch0 = tid / 96;     // 0 (even channels) or 1 (odd channels)
            const float* g0 = src + ((size_t)(c0 + ch0) * NH + r) * NW + 2 * u;
            float*       l0 = &tile[(ch0 * 4 + rw) * RS + 2 * u];
#pragma unroll
            for (int i = 0; i < 24; ++i) {
                const float* g = g0 + (size_t)i * (2 * NH * NW);  // +2 channels
                float*       l = l0 + i * (8 * RS);               // +2 channels in LDS
#if ASYNC_LDS
                __builtin_amdgcn_global_load_async_to_lds_b64(
                    (as1_v2i)(uintptr_t)g,
                    (as3_v2i)(uint32_t)(uintptr_t)l,  // low 32 bits of flat addr = LDS offset
                    0, 0);
#else
                l[0] = g[0];
                l[1] = g[1];
#endif
            }
        }
    }
#if ASYNC_LDS
    __builtin_amdgcn_s_wait_asynccnt(0);
#endif
    __syncthreads();

    // ---- Consume: thread = (channel cl, tap-row k); slide a 4-wide column window over x.
    // Window invariant at iteration x: w0..w3 = input cols x-2..x+1 (0 when out of range).
    const int  k   = tid & 3;
    const bool k3  = (k == 3);               // masked taps: (i==3, j>=2)
    const bool rok = (y + k - 3) >= 0;       // input row in range
    const int  base = tid * RS;              // (cl*4+k)*RS == tid*RS

    vf4* outp = (vf4*)out + (size_t)(b * NH + y) * NW * (NC * 4) + c0 * 4 + tid;

    float w0 = 0.0f, w1 = 0.0f;
    float w2 = tile[base + 0];
    float w3 = tile[base + 1];

    for (int x = 0; x < NW; ++x) {
        vf4 q;
        q.x = rok ? w0 : 0.0f;
        q.y = rok ? w1 : 0.0f;
        q.z = (rok && !k3) ? w2 : 0.0f;
        q.w = (rok && !k3) ? w3 : 0.0f;
        // output written exactly once, never re-read -> non-temporal b128 store;
        // consecutive lanes store consecutive 16B -> 512B contiguous per wave
        __builtin_nontemporal_store(q, outp + (size_t)x * (NC * 4));
        w0 = w1; w1 = w2; w2 = w3;
        const int nc = x + 2;                // column needed for next iteration
        w3 = (nc < NW) ? tile[base + nc] : 0.0f;
    }
}

extern "C" void kernel_launch(void* const* d_in, const int* in_sizes, int n_in,
                              void* d_out, int out_size, void* d_ws, size_t ws_size,
                              hipStream_t stream) {
    (void)in_sizes; (void)n_in; (void)d_ws; (void)ws_size; (void)out_size;
    const float* in  = (const float*)d_in[0];
    float*       out = (float*)d_out;
    dim3 grid(NCG, NH, NB);   // 4 x 48 x 8 = 1536 blocks
    hipLaunchKernelGGL(onehot_tap_gather, grid, dim3(192), 0, stream, in, out);
}